// BSplineLoss_82214263980150
// MI455X (gfx1250) — compile-verified
//
#include <hip/hip_runtime.h>
#include <cstdint>

// Problem constants (from the reference)
constexpr int NN = 64;          // knots / control points per sequence
constexpr int NS = 20;          // samples
constexpr int DEG = 3;
constexpr int BLK = 64;         // threads per block (2 wave32 waves)

// ---------------------------------------------------------------------------
// De Boor evaluation for one sequence of one batch element.
//   kn : LDS pointer to this thread's 64 knots
//   g  : global pointer to seq[b][0][0]  (layout [64][3] = knot,cx,cy)
//   out: 20 sampled (x,y) points
// ---------------------------------------------------------------------------
__device__ __forceinline__ float kf_at(const float* __restrict__ kn, int i) {
  // kf = [0,0,0,0, knots...]
  return (i < DEG + 1) ? 0.0f : kn[i - (DEG + 1)];
}

__device__ void eval_seq(const float* __restrict__ kn,
                         const float* __restrict__ g,
                         float2* __restrict__ out) {
  const float high = kn[60];          // kf[64] = knots[60]; low = kf[3] = 0
  int cnt = 1;                        // searchsorted_right count; inner[0]=0 <= t always
  for (int j = 0; j < NS; ++j) {
    const float t = ((float)j * (1.0f / 19.0f)) * high;
    // inner[i] = (i==0 ? 0 : knots[i-1]), i in [0,61); t is increasing -> walk forward
    while (cnt < 61 && kn[cnt - 1] <= t) ++cnt;
    const int i0 = cnt - 1;           // s - DEG, with s = cnt + DEG - 1

    float px[4], py[4], pw[4];
#pragma unroll
    for (int k = 0; k < 4; ++k) {
      const float* cp = g + (size_t)(i0 + k) * 3;
      px[k] = cp[1];
      py[k] = cp[2];
      pw[k] = 1.0f;                   // homogeneous weight channel (matches reference)
    }
#pragma unroll
    for (int l = 1; l <= DEG; ++l) {
      float nx[4], ny[4], nw[4];
#pragma unroll
      for (int k = 3; k >= 1; --k) {
        if (k < l) continue;          // reference: p updated only where karr >= l
        const int idx = i0 + k;
        const float ki = kf_at(kn, idx);
        const float kj = kf_at(kn, idx + DEG + 1 - l);
        const float denom = kj - ki;
        const float alpha = (denom != 0.0f) ? (t - ki) / denom : 0.0f;
        const float oma = 1.0f - alpha;
        nx[k] = oma * px[k - 1] + alpha * px[k];
        ny[k] = oma * py[k - 1] + alpha * py[k];
        nw[k] = oma * pw[k - 1] + alpha * pw[k];
      }
#pragma unroll
      for (int k = 1; k <= 3; ++k) {
        if (k < l) continue;
        px[k] = nx[k]; py[k] = ny[k]; pw[k] = nw[k];
      }
    }
    out[j].x = px[3] / pw[3];
    out[j].y = py[3] / pw[3];
  }
}

// ---------------------------------------------------------------------------
// Main kernel: one thread per batch element. Knots staged to LDS via the
// CDNA5 async global->LDS path (ASYNCcnt), span search + kf lookups served
// from LDS, coefficients from global (cache-resident). Per-block partial sum
// written to d_ws (deterministic wave32 shuffle reduction).
// ---------------------------------------------------------------------------
__global__ __launch_bounds__(BLK) void bspline_loss_main(
    const float* __restrict__ pred, const float* __restrict__ tru,
    float* __restrict__ partials) {
  __shared__ float sk[2][BLK][NN + 1];   // +1 pad -> conflict-free banks
  __shared__ float sw[BLK / 32];

  const int tid = threadIdx.x;
  const size_t b = (size_t)blockIdx.x * BLK + tid;
  const float* gp = pred + b * (size_t)(NN * 3);
  const float* gt = tru  + b * (size_t)(NN * 3);

  // Stage knots (stride-12B gather) into LDS with async loads; one s_wait.
#pragma unroll 4
  for (int n = 0; n < NN; ++n) {
    const unsigned lp = (unsigned)(uintptr_t)&sk[0][tid][n];
    const unsigned lt = (unsigned)(uintptr_t)&sk[1][tid][n];
    const unsigned long long ap = (unsigned long long)(uintptr_t)(gp + 3 * n);
    const unsigned long long at = (unsigned long long)(uintptr_t)(gt + 3 * n);
    asm volatile("global_load_async_to_lds_b32 %0, %1, off"
                 :: "v"(lp), "v"(ap) : "memory");
    asm volatile("global_load_async_to_lds_b32 %0, %1, off"
                 :: "v"(lt), "v"(at) : "memory");
  }
  asm volatile("s_wait_asynccnt 0" ::: "memory");

  float2 ts[NS], ps[NS];
  eval_seq(&sk[1][tid][0], gt, ts);
  eval_seq(&sk[0][tid][0], gp, ps);

  float acc = 0.0f;
#pragma unroll
  for (int j = 0; j < NS; ++j) {
    const float dx = ps[j].x - ts[j].x;
    const float dy = ps[j].y - ts[j].y;
    acc += dx * dx + dy * dy;
  }

  // wave32 reduction, then combine the block's two waves (fixed order)
  for (int off = 16; off > 0; off >>= 1) acc += __shfl_down(acc, off, 32);
  if ((tid & 31) == 0) sw[tid >> 5] = acc;
  __syncthreads();
  if (tid == 0) partials[blockIdx.x] = sw[0] + sw[1];
}

// ---------------------------------------------------------------------------
// Final reduction on one wave using V_WMMA_F32_16X16X4_F32.
// A = ones(16x4)  =>  D[m,n] = sum_k B[k,n] + C[m,n]  (all rows identical).
// Chaining C accumulates 64 values per WMMA independent of B's lane layout;
// sum of c[0] over all 32 lanes == 2 * grand total.
// EXEC stays all-1s: out-of-range lanes are clamped+zeroed branchlessly.
// ---------------------------------------------------------------------------
typedef float v2f __attribute__((ext_vector_type(2)));
typedef float v8f __attribute__((ext_vector_type(8)));

__global__ __launch_bounds__(32) void bspline_loss_reduce(
    const float* __restrict__ partials, float* __restrict__ out,
    int n, float scale) {
  const int lane = threadIdx.x;
  v8f c = {0.f, 0.f, 0.f, 0.f, 0.f, 0.f, 0.f, 0.f};
  v2f a = {1.0f, 1.0f};
  for (int base = 0; base < n; base += 64) {
    const int i = base + lane * 2;
    const int i0 = (i     < n) ? i     : (n - 1);
    const int i1 = (i + 1 < n) ? i + 1 : (n - 1);
    const float m0 = (i     < n) ? 1.0f : 0.0f;
    const float m1 = (i + 1 < n) ? 1.0f : 0.0f;
    v2f bm;
    bm[0] = partials[i0] * m0;
    bm[1] = partials[i1] * m1;
    c = __builtin_amdgcn_wmma_f32_16x16x4_f32(
        /*neg_a=*/false, a, /*neg_b=*/false, bm,
        /*c_mod=*/(short)0, c, /*reuse_a=*/false, /*reuse_b=*/false);
  }
  float r = c[0];
  for (int off = 16; off > 0; off >>= 1) r += __shfl_down(r, off, 32);
  if (lane == 0) out[0] = r * 0.5f * scale;
}

// ---------------------------------------------------------------------------
extern "C" void kernel_launch(void* const* d_in, const int* in_sizes, int n_in,
                              void* d_out, int out_size, void* d_ws, size_t ws_size,
                              hipStream_t stream) {
  const float* pred = (const float*)d_in[0];
  const float* tru  = (const float*)d_in[1];
  // d_in[2] (true_masks) is unused: the reference rebuilds mask = ones.

  const int Bt   = in_sizes[0] / (NN * 3);   // 65536
  const int nblk = Bt / BLK;                 // 1024 (Bt divisible by BLK here)
  float* partials = (float*)d_ws;            // nblk * 4 bytes of scratch

  bspline_loss_main<<<nblk, BLK, 0, stream>>>(pred, tru, partials);

  const float scale = 1.0f / ((float)Bt * (float)NS);  // / mask.sum()
  bspline_loss_reduce<<<1, 32, 0, stream>>>(partials, (float*)d_out, nblk, scale);
}